// ModelName_11656541241545
// MI455X (gfx1250) — compile-verified
//
#include <hip/hip_runtime.h>
#include <hip/hip_bf16.h>
#include <math.h>

// ---------------------------------------------------------------------------
// 2-layer GAT (heads=1) + linear classifier + log_softmax, fp32 end-to-end.
// GEMMs use V_WMMA_F32_16X16X4_F32 (fp32 matrix pipe, wave32 layouts per ISA).
// Edge softmax/aggregation uses hw float atomics (L2-resident on 192MB L2).
// ---------------------------------------------------------------------------

typedef __attribute__((ext_vector_type(2))) float v2f;
typedef __attribute__((ext_vector_type(8))) float v8f;

#define GAT_H 64
#define GAT_C 40
#define NEG_SLOPE 0.2f

// monotone float <-> ordered-int encoding (signed compare preserves float order)
__device__ __forceinline__ int enc_ord(float f) {
    int i = __float_as_int(f);
    return (i >= 0) ? i : (i ^ 0x7fffffff);
}
__device__ __forceinline__ float dec_ord(int i) {
    return __int_as_float((i >= 0) ? i : (i ^ 0x7fffffff));
}
#define ENC_NEG_INF ((int)0x807FFFFF)  // enc_ord(-inf)

__device__ __forceinline__ float lrelu(float x) {
    return (x > 0.0f) ? x : NEG_SLOPE * x;
}

__device__ __forceinline__ void edge_sd(int id, const int* __restrict__ src,
                                        const int* __restrict__ dst, int E,
                                        int& s, int& d) {
    if (id < E) { s = src[id]; d = dst[id]; }
    else        { s = id - E;  d = id - E;  }   // appended self-loops
}

// ---------------------------------------------------------------------------
// GEMM: Hout[N,64] = X[N,K] @ W[K,64]   (K = 128 or 64; N multiple of 16)
// One wave per 16-row strip; 4 accumulators cover all 64 output columns.
// WMMA layouts (ISA 7.12.2):
//   A 16x4 f32 : lane<16  -> {A[M=l][k0+0], A[M=l][k0+1]}
//                lane>=16 -> {A[M=l][k0+2], A[M=l][k0+3]}
//   B 4x16 f32 : lane<16  -> {B[k0+0][n],  B[k0+1][n]}
//                lane>=16 -> {B[k0+2][n],  B[k0+3][n]}
//   C/D 16x16  : vgpr r, lane<16 -> (M=r,   N=l); lane>=16 -> (M=8+r, N=l)
// ---------------------------------------------------------------------------
__global__ void gat_gemm_wmma(const float* __restrict__ X,
                              const float* __restrict__ W,
                              float* __restrict__ Hout,
                              int N, int K) {
    const int lane = threadIdx.x & 31;
    const int wave = threadIdx.x >> 5;
    const int row0 = (blockIdx.x * 8 + wave) * 16;
    if (row0 >= N) return;                       // wave-uniform branch
    const int half = lane >> 4;                  // 0: lanes 0-15, 1: 16-31
    const int l16  = lane & 15;

    v8f acc0 = {}, acc1 = {}, acc2 = {}, acc3 = {};
    const float* xrow = X + (size_t)(row0 + l16) * (size_t)K;

    for (int k0 = 0; k0 < K; k0 += 4) {
        v2f a;
        a.x = xrow[k0 + half * 2 + 0];
        a.y = xrow[k0 + half * 2 + 1];
        const float* wr0 = W + (size_t)(k0 + half * 2 + 0) * GAT_H;
        const float* wr1 = W + (size_t)(k0 + half * 2 + 1) * GAT_H;
        v2f b0, b1, b2, b3;
        b0.x = wr0[l16 +  0]; b0.y = wr1[l16 +  0];
        b1.x = wr0[l16 + 16]; b1.y = wr1[l16 + 16];
        b2.x = wr0[l16 + 32]; b2.y = wr1[l16 + 32];
        b3.x = wr0[l16 + 48]; b3.y = wr1[l16 + 48];
        acc0 = __builtin_amdgcn_wmma_f32_16x16x4_f32(false, a, false, b0, (short)0, acc0, false, false);
        acc1 = __builtin_amdgcn_wmma_f32_16x16x4_f32(false, a, false, b1, (short)0, acc1, false, false);
        acc2 = __builtin_amdgcn_wmma_f32_16x16x4_f32(false, a, false, b2, (short)0, acc2, false, false);
        acc3 = __builtin_amdgcn_wmma_f32_16x16x4_f32(false, a, false, b3, (short)0, acc3, false, false);
    }

#pragma unroll
    for (int r = 0; r < 8; ++r) {
        int row = row0 + half * 8 + r;
        float* o = Hout + (size_t)row * GAT_H + l16;
        o[0]  = acc0[r];
        o[16] = acc1[r];
        o[32] = acc2[r];
        o[48] = acc3[r];
    }
}

// alpha_s[n] = h[n,:].a_src ; alpha_d[n] = h[n,:].a_dst  (wave per node)
__global__ void gat_alphas(float* __restrict__ as_out, float* __restrict__ ad_out,
                           const float* __restrict__ h,
                           const float* __restrict__ a_src,
                           const float* __restrict__ a_dst, int N) {
    int node = (blockIdx.x * blockDim.x + threadIdx.x) >> 5;
    int lane = threadIdx.x & 31;
    if (node >= N) return;
    const float* hr = h + (size_t)node * GAT_H;
    float h0 = hr[lane], h1 = hr[lane + 32];
    float s = h0 * a_src[lane] + h1 * a_src[lane + 32];
    float d = h0 * a_dst[lane] + h1 * a_dst[lane + 32];
#pragma unroll
    for (int m = 16; m >= 1; m >>= 1) {
        s += __shfl_xor(s, m, 32);
        d += __shfl_xor(d, m, 32);
    }
    if (lane == 0) { as_out[node] = s; ad_out[node] = d; }
}

// zero agg[N*64], set mEnc = enc(-inf), denom = 0
__global__ void gat_init(float* __restrict__ agg, int* __restrict__ mEnc,
                         float* __restrict__ denom, int N) {
    int tid = blockIdx.x * blockDim.x + threadIdx.x;
    if (tid < N * GAT_H) agg[tid] = 0.0f;
    if (tid < N) { mEnc[tid] = ENC_NEG_INF; denom[tid] = 0.0f; }
}

// pass 1: segment max of e over dst (ordered-int atomicMax)
__global__ void gat_edge_max(int* __restrict__ mEnc,
                             const float* __restrict__ as, const float* __restrict__ ad,
                             const int* __restrict__ src, const int* __restrict__ dst,
                             int E, int total) {
    int id = blockIdx.x * blockDim.x + threadIdx.x;
    if (id >= total) return;
    int s, d; edge_sd(id, src, dst, E, s, d);
    float e = lrelu(as[s] + ad[d]);
    atomicMax(&mEnc[d], enc_ord(e));
}

// pass 2: denom[d] += exp(e - m[d])
__global__ void gat_edge_sum(float* __restrict__ denom,
                             const int* __restrict__ mEnc,
                             const float* __restrict__ as, const float* __restrict__ ad,
                             const int* __restrict__ src, const int* __restrict__ dst,
                             int E, int total) {
    int id = blockIdx.x * blockDim.x + threadIdx.x;
    if (id >= total) return;
    int s, d; edge_sd(id, src, dst, E, s, d);
    float e = lrelu(as[s] + ad[d]);
    float ex = __expf(e - dec_ord(mEnc[d]));
    atomicAdd(&denom[d], ex);
}

// pass 3: agg[d,:] += h[s,:] * w   (wave per edge, 2 feats/lane)
__global__ void gat_edge_aggregate(float* __restrict__ agg,
                                   const float* __restrict__ h,
                                   const int* __restrict__ mEnc,
                                   const float* __restrict__ denom,
                                   const float* __restrict__ as, const float* __restrict__ ad,
                                   const int* __restrict__ src, const int* __restrict__ dst,
                                   int E, int total) {
    int wid  = (blockIdx.x * blockDim.x + threadIdx.x) >> 5;
    int lane = threadIdx.x & 31;
    if (wid >= total) return;                    // wave-uniform
    int s, d; edge_sd(wid, src, dst, E, s, d);
    float e  = lrelu(as[s] + ad[d]);
    float ex = __expf(e - dec_ord(mEnc[d]));
    float w  = ex / (denom[d] + 1e-16f);
    const float* hs = h + (size_t)s * GAT_H;
    float* o = agg + (size_t)d * GAT_H;
    atomicAdd(&o[lane],      hs[lane]      * w);
    atomicAdd(&o[lane + 32], hs[lane + 32] * w);
}

// buf[n,f] = act(buf[n,f] + b[f])
__global__ void gat_bias_act(float* __restrict__ buf, const float* __restrict__ b,
                             int N, int do_relu) {
    int tid = blockIdx.x * blockDim.x + threadIdx.x;
    if (tid >= N * GAT_H) return;
    float v = buf[tid] + b[tid & (GAT_H - 1)];
    if (do_relu) v = fmaxf(v, 0.0f);
    buf[tid] = v;
}

// logits[n,c] = z[n,:].Wl[:,c] + bl[c]
__global__ void gat_logits(float* __restrict__ logits, const float* __restrict__ z,
                           const float* __restrict__ Wl, const float* __restrict__ bl,
                           int N) {
    int tid = blockIdx.x * blockDim.x + threadIdx.x;
    if (tid >= N * GAT_C) return;
    int n = tid / GAT_C, c = tid - n * GAT_C;
    const float* zr = z + (size_t)n * GAT_H;
    float acc = bl[c];
#pragma unroll
    for (int k = 0; k < GAT_H; ++k) acc = fmaf(zr[k], Wl[k * GAT_C + c], acc);
    logits[tid] = acc;
}

// row-wise log_softmax over C=40 (wave per node)
__global__ void gat_logsoftmax(float* __restrict__ out, const float* __restrict__ logits,
                               int N) {
    int node = (blockIdx.x * blockDim.x + threadIdx.x) >> 5;
    int lane = threadIdx.x & 31;
    if (node >= N) return;
    const float* lr = logits + (size_t)node * GAT_C;
    float v0 = lr[lane];
    float v1 = (lane < GAT_C - 32) ? lr[lane + 32] : -INFINITY;
    float mx = fmaxf(v0, v1);
#pragma unroll
    for (int m = 16; m >= 1; m >>= 1) mx = fmaxf(mx, __shfl_xor(mx, m, 32));
    float se = __expf(v0 - mx) + ((lane < GAT_C - 32) ? __expf(v1 - mx) : 0.0f);
#pragma unroll
    for (int m = 16; m >= 1; m >>= 1) se += __shfl_xor(se, m, 32);
    float lse = mx + logf(se);
    float* orow = out + (size_t)node * GAT_C;
    orow[lane] = v0 - lse;
    if (lane < GAT_C - 32) orow[lane + 32] = v1 - lse;
}

// ---------------------------------------------------------------------------
extern "C" void kernel_launch(void* const* d_in, const int* in_sizes, int n_in,
                              void* d_out, int out_size, void* d_ws, size_t ws_size,
                              hipStream_t stream) {
    const int F_IN = 128;
    const int N = in_sizes[0] / F_IN;
    const int E = in_sizes[1] / 2;
    const int TOT = E + N;                 // edges incl. self-loops

    const float* x      = (const float*)d_in[0];
    const int*   eidx   = (const int*)  d_in[1];
    const float* W1     = (const float*)d_in[2];
    const float* a_src1 = (const float*)d_in[3];
    const float* a_dst1 = (const float*)d_in[4];
    const float* b1     = (const float*)d_in[5];
    const float* W2     = (const float*)d_in[6];
    const float* a_src2 = (const float*)d_in[7];
    const float* a_dst2 = (const float*)d_in[8];
    const float* b2     = (const float*)d_in[9];
    const float* Wl     = (const float*)d_in[10];
    const float* bl     = (const float*)d_in[11];
    const int* src = eidx;
    const int* dst = eidx + E;

    // workspace carve-up (256B aligned)
    char* ws = (char*)d_ws;
    size_t off = 0;
    auto carve = [&](size_t bytes) {
        void* p = ws + off;
        off += (bytes + 255) & ~(size_t)255;
        return p;
    };
    float* bufA   = (float*)carve((size_t)N * GAT_H * 4);  // h (per layer)
    float* bufB   = (float*)carve((size_t)N * GAT_H * 4);  // agg / z (per layer)
    float* bufC   = (float*)carve((size_t)N * GAT_C * 4);  // logits
    float* alphaS = (float*)carve((size_t)N * 4);
    float* alphaD = (float*)carve((size_t)N * 4);
    int*   mEnc   = (int*)  carve((size_t)N * 4);
    float* denom  = (float*)carve((size_t)N * 4);
    (void)ws_size;

    const int BLK = 256;
    const int gGemm  = (N / 16 + 7) / 8;            // 8 waves/block, 16 rows/wave
    const int gWaveN = (N * 32 + BLK - 1) / BLK;    // wave-per-node kernels
    const int gNH    = (N * GAT_H + BLK - 1) / BLK;
    const int gEdge  = (TOT + BLK - 1) / BLK;
    const int gEdgeW = ((size_t)TOT * 32 + BLK - 1) / BLK;  // wave-per-edge
    const int gNC    = (N * GAT_C + BLK - 1) / BLK;

    // ---------------- layer 1 ----------------
    gat_gemm_wmma<<<gGemm, BLK, 0, stream>>>(x, W1, bufA, N, F_IN);
    gat_alphas<<<gWaveN, BLK, 0, stream>>>(alphaS, alphaD, bufA, a_src1, a_dst1, N);
    gat_init<<<gNH, BLK, 0, stream>>>(bufB, mEnc, denom, N);
    gat_edge_max<<<gEdge, BLK, 0, stream>>>(mEnc, alphaS, alphaD, src, dst, E, TOT);
    gat_edge_sum<<<gEdge, BLK, 0, stream>>>(denom, mEnc, alphaS, alphaD, src, dst, E, TOT);
    gat_edge_aggregate<<<gEdgeW, BLK, 0, stream>>>(bufB, bufA, mEnc, denom,
                                                   alphaS, alphaD, src, dst, E, TOT);
    gat_bias_act<<<gNH, BLK, 0, stream>>>(bufB, b1, N, 1);   // +b1, relu -> z1

    // ---------------- layer 2 ----------------
    gat_gemm_wmma<<<gGemm, BLK, 0, stream>>>(bufB, W2, bufA, N, GAT_H);
    gat_alphas<<<gWaveN, BLK, 0, stream>>>(alphaS, alphaD, bufA, a_src2, a_dst2, N);
    gat_init<<<gNH, BLK, 0, stream>>>(bufB, mEnc, denom, N);
    gat_edge_max<<<gEdge, BLK, 0, stream>>>(mEnc, alphaS, alphaD, src, dst, E, TOT);
    gat_edge_sum<<<gEdge, BLK, 0, stream>>>(denom, mEnc, alphaS, alphaD, src, dst, E, TOT);
    gat_edge_aggregate<<<gEdgeW, BLK, 0, stream>>>(bufB, bufA, mEnc, denom,
                                                   alphaS, alphaD, src, dst, E, TOT);
    gat_bias_act<<<gNH, BLK, 0, stream>>>(bufB, b2, N, 0);   // +b2 -> z2

    // ---------------- classifier + log_softmax ----------------
    gat_logits<<<gNC, BLK, 0, stream>>>(bufC, bufB, Wl, bl, N);
    gat_logsoftmax<<<gWaveN, BLK, 0, stream>>>((float*)d_out, bufC, N);
}